// Attention_35613868819051
// MI455X (gfx1250) — compile-verified
//
#include <hip/hip_runtime.h>
#include <math.h>

typedef __attribute__((ext_vector_type(2))) float v2f;
typedef __attribute__((ext_vector_type(8))) float v8f;

#define BB 16
#define NN 4096
#define CC 1024
#define HH 16
#define DH 64   // CC/HH

// ---------------------------------------------------------------------------
// Small VALU kernels (all trivially cheap: ~17M MACs each)
// ---------------------------------------------------------------------------

// q[b][c'] = sum_c x[b,0,c] * Wq[c',c]
__global__ void qproj_kernel(const float* __restrict__ x, const float* __restrict__ Wq,
                             float* __restrict__ q) {
  int b = blockIdx.y;
  int cp = blockIdx.x * blockDim.x + threadIdx.x;
  const float* xr = x + (size_t)b * NN * CC;        // token 0 row
  const float* wr = Wq + (size_t)cp * CC;
  float acc = 0.f;
  for (int c = 0; c < CC; ++c) acc = fmaf(xr[c], wr[c], acc);
  q[b * CC + cp] = acc;
}

// wk_eff[b][h][c] = scale * sum_d q[b][h*DH+d] * Wk[h*DH+d][c]
__global__ void fold_k_kernel(const float* __restrict__ q, const float* __restrict__ Wk,
                              float* __restrict__ wk_eff, float scale) {
  int b = blockIdx.y;
  int idx = blockIdx.x * blockDim.x + threadIdx.x;  // h*CC + c
  int h = idx / CC;
  int c = idx - h * CC;
  float acc = 0.f;
  const float* qh = q + b * CC + h * DH;
  const float* wk = Wk + (size_t)(h * DH) * CC + c;
  for (int d = 0; d < DH; ++d) acc = fmaf(qh[d], wk[(size_t)d * CC], acc);
  wk_eff[((size_t)b * HH + h) * CC + c] = acc * scale;
}

// ---------------------------------------------------------------------------
// Pass 1: scores s[b][h][n] = x[b,n,:] . wk_eff[b,h,:]  (+ additive mask)
// One wave32 per 16-token tile. WMMA f32 16x16x4: A = x (16 tok x 4 ch),
// B = wk_eff^T (4 ch x 16 heads), 256 K-steps over C.
// ---------------------------------------------------------------------------
__global__ __launch_bounds__(32) void scores_wmma_kernel(
    const float* __restrict__ x, const float* __restrict__ wk_eff,
    const float* __restrict__ mask, float* __restrict__ s) {
  const int ntiles = NN >> 4;
  const int b  = blockIdx.x / ntiles;
  const int n0 = (blockIdx.x - b * ntiles) << 4;
  const int lane = threadIdx.x;
  const int m15 = lane & 15;   // A: token row ; B: head column
  const int kh  = lane >> 4;   // selects K pair {2kh, 2kh+1}

  const float* xrow = x      + ((size_t)(b * NN + n0 + m15)) * CC + 2 * kh;
  const float* brow = wk_eff + ((size_t)(b * HH + m15)) * CC + 2 * kh;

  v8f acc = {};
  for (int k0 = 0; k0 < CC; k0 += 4) {
    v2f a, bm;
    a.x  = xrow[k0];  a.y  = xrow[k0 + 1];
    bm.x = brow[k0];  bm.y = brow[k0 + 1];
    acc = __builtin_amdgcn_wmma_f32_16x16x4_f32(false, a, false, bm,
                                                (short)0, acc, false, false);
  }
  // D layout: acc[r] = S[token = r + 8*kh][head = m15]
  for (int r = 0; r < 8; ++r) {
    int n = n0 + r + 8 * kh;
    float madd = (n == 0) ? 0.f : mask[(size_t)b * (NN - 1) + (n - 1)];
    s[((size_t)(b * HH + m15)) * NN + n] = acc[r] + madd;
  }
}

// ---------------------------------------------------------------------------
// Softmax over N per (b,h) row, in place. One 256-thread block per row.
// ---------------------------------------------------------------------------
__global__ void softmax_kernel(float* __restrict__ s) {
  __shared__ float red[256];
  float* row = s + (size_t)blockIdx.x * NN;
  int tid = threadIdx.x;

  float mx = -INFINITY;
  for (int i = tid; i < NN; i += 256) mx = fmaxf(mx, row[i]);
  red[tid] = mx; __syncthreads();
  for (int off = 128; off > 0; off >>= 1) {
    if (tid < off) red[tid] = fmaxf(red[tid], red[tid + off]);
    __syncthreads();
  }
  mx = red[0]; __syncthreads();

  float sum = 0.f;
  for (int i = tid; i < NN; i += 256) {
    float e = expf(row[i] - mx);
    row[i] = e;
    sum += e;
  }
  red[tid] = sum; __syncthreads();
  for (int off = 128; off > 0; off >>= 1) {
    if (tid < off) red[tid] += red[tid + off];
    __syncthreads();
  }
  float inv = 1.f / red[0];
  for (int i = tid; i < NN; i += 256) row[i] *= inv;
}

// ---------------------------------------------------------------------------
// Pass 2: t[b][h][c] = sum_n p[b,h,n] * x[b,n,c]
// One wave32 per 16-channel tile. WMMA: A = p (16 heads x 4 tok),
// B = x (4 tok x 16 ch), 1024 K-steps over N.
// ---------------------------------------------------------------------------
__global__ __launch_bounds__(32) void wsum_wmma_kernel(
    const float* __restrict__ p, const float* __restrict__ x,
    float* __restrict__ t) {
  const int ctiles = CC >> 4;
  const int b  = blockIdx.x / ctiles;
  const int c0 = (blockIdx.x - b * ctiles) << 4;
  const int lane = threadIdx.x;
  const int m15 = lane & 15;   // A: head row ; B: channel column
  const int kh  = lane >> 4;

  const float* prow = p + ((size_t)(b * HH + m15)) * NN + 2 * kh;
  const float* xcol = x + ((size_t)b * NN) * CC + c0 + m15;

  v8f acc = {};
  for (int k0 = 0; k0 < NN; k0 += 4) {
    v2f a, bm;
    a.x  = prow[k0];  a.y = prow[k0 + 1];
    bm.x = xcol[(size_t)(k0 + 2 * kh)     * CC];
    bm.y = xcol[(size_t)(k0 + 2 * kh + 1) * CC];
    acc = __builtin_amdgcn_wmma_f32_16x16x4_f32(false, a, false, bm,
                                                (short)0, acc, false, false);
  }
  // D layout: acc[r] = T[head = r + 8*kh][channel = c0 + m15]
  for (int r = 0; r < 8; ++r) {
    int h = r + 8 * kh;
    t[((size_t)(b * HH + h)) * CC + c0 + m15] = acc[r];
  }
}

// ---------------------------------------------------------------------------
// xcls[b][h*DH+d] = sum_c Wv[h*DH+d][c] * t[b][h][c]
// ---------------------------------------------------------------------------
__global__ void vproj_kernel(const float* __restrict__ t, const float* __restrict__ Wv,
                             float* __restrict__ xcls) {
  int b = blockIdx.y;
  int cp = blockIdx.x * blockDim.x + threadIdx.x;   // h*DH + d
  int h = cp / DH;
  const float* wr = Wv + (size_t)cp * CC;
  const float* tr = t + ((size_t)b * HH + h) * CC;
  float acc = 0.f;
  for (int c = 0; c < CC; ++c) acc = fmaf(wr[c], tr[c], acc);
  xcls[b * CC + cp] = acc;
}

// out[b][c'] = bp[c'] + sum_c xcls[b][c] * Wp[c'][c]
__global__ void oproj_kernel(const float* __restrict__ xcls, const float* __restrict__ Wp,
                             const float* __restrict__ bp, float* __restrict__ out) {
  int b = blockIdx.y;
  int cp = blockIdx.x * blockDim.x + threadIdx.x;
  const float* xr = xcls + (size_t)b * CC;
  const float* wr = Wp + (size_t)cp * CC;
  float acc = bp[cp];
  for (int c = 0; c < CC; ++c) acc = fmaf(xr[c], wr[c], acc);
  out[b * CC + cp] = acc;
}

// ---------------------------------------------------------------------------

extern "C" void kernel_launch(void* const* d_in, const int* in_sizes, int n_in,
                              void* d_out, int out_size, void* d_ws, size_t ws_size,
                              hipStream_t stream) {
  const float* x    = (const float*)d_in[0];
  const float* mask = (const float*)d_in[1];
  const float* Wq   = (const float*)d_in[2];
  const float* Wk   = (const float*)d_in[3];
  const float* Wv   = (const float*)d_in[4];
  const float* Wp   = (const float*)d_in[5];
  const float* bp   = (const float*)d_in[6];
  float* out = (float*)d_out;

  float* ws = (float*)d_ws;
  float* q      = ws;                                   // B*C          = 16384
  float* wk_eff = q + BB * CC;                          // B*H*C        = 262144
  float* s      = wk_eff + (size_t)BB * HH * CC;        // B*H*N        = 1048576 (softmax in place)
  float* t      = s + (size_t)BB * HH * NN;             // B*H*C        = 262144
  float* xcls   = t + (size_t)BB * HH * CC;             // B*C          = 16384

  const float scale = 0.125f;   // (C/H)^-0.5 = 64^-0.5

  qproj_kernel<<<dim3(CC / 256, BB), 256, 0, stream>>>(x, Wq, q);
  fold_k_kernel<<<dim3(HH * CC / 256, BB), 256, 0, stream>>>(q, Wk, wk_eff, scale);
  scores_wmma_kernel<<<BB * (NN / 16), 32, 0, stream>>>(x, wk_eff, mask, s);
  softmax_kernel<<<BB * HH, 256, 0, stream>>>(s);
  wsum_wmma_kernel<<<BB * (CC / 16), 32, 0, stream>>>(s, x, t);
  vproj_kernel<<<dim3(CC / 256, BB), 256, 0, stream>>>(t, Wv, xcls);
  oproj_kernel<<<dim3(CC / 256, BB), 256, 0, stream>>>(xcls, Wp, bp, out);
}